// InterclassLoss_70531952935127
// MI455X (gfx1250) — compile-verified
//
#include <hip/hip_runtime.h>

// ext_vector types so __builtin_nontemporal_load accepts them (HIP float4 is a struct).
typedef float v4f __attribute__((ext_vector_type(4)));
typedef int   v4i __attribute__((ext_vector_type(4)));

#define NCLS 5
#define BLOCKS 1024
#define THREADS 256

// Per-sample loss term: weight * (-p_true), with
//   weight = (|argmax - t| + 1) / M(t),  M(t) = 11 + (t-2)^2  (exact for t in 0..4)
__device__ __forceinline__ float sample_term(const float p[NCLS], int t) {
    // argmax, first max wins (matches jnp.argmax tie rule)
    float best = p[0];
    int   idx  = 0;
#pragma unroll
    for (int j = 1; j < NCLS; ++j) {
        if (p[j] > best) { best = p[j]; idx = j; }
    }
    // gather p[t] via compare/select (t is 0..4)
    float pt = p[0];
#pragma unroll
    for (int j = 1; j < NCLS; ++j) {
        if (t == j) pt = p[j];
    }
    int   ad  = idx - t; if (ad < 0) ad = -ad;          // |argmax - t|
    float num = (float)(ad + 1);
    // 1/M(t): M in {15,12,11,12,15} for t in {0..4}
    float rM = (t == 2) ? (1.0f / 11.0f)
             : ((t == 1 || t == 3) ? (1.0f / 12.0f) : (1.0f / 15.0f));
    return -(num * rM) * pt;
}

__global__ void __launch_bounds__(THREADS)
interclass_partial_kernel(const float* __restrict__ pred,
                          const long long* __restrict__ truth,
                          float* __restrict__ partial,
                          long long n) {
    const long long groups = n >> 2;                 // 4 samples per group
    const long long tid    = (long long)blockIdx.x * blockDim.x + threadIdx.x;
    const long long stride = (long long)gridDim.x * blockDim.x;

    const v4f* __restrict__ pred4 = (const v4f*)pred;    // 5 x v4f per group (80B, 16B aligned)
    const v4i* __restrict__ tr4   = (const v4i*)truth;   // 2 x v4i per group (4 x int64)

    float sum = 0.0f;

    for (long long g = tid; g < groups; g += stride) {
        // 5x global_load_b128 th:NT -> 20 contiguous floats = 4 samples' logits
        v4f v0 = __builtin_nontemporal_load(pred4 + 5 * g + 0);
        v4f v1 = __builtin_nontemporal_load(pred4 + 5 * g + 1);
        v4f v2 = __builtin_nontemporal_load(pred4 + 5 * g + 2);
        v4f v3 = __builtin_nontemporal_load(pred4 + 5 * g + 3);
        v4f v4 = __builtin_nontemporal_load(pred4 + 5 * g + 4);
        // 2x global_load_b128 th:NT -> 4 int64 truths (low dwords are the values)
        v4i t01 = __builtin_nontemporal_load(tr4 + 2 * g + 0);
        v4i t23 = __builtin_nontemporal_load(tr4 + 2 * g + 1);

        float p[4][NCLS] = {
            { v0.x, v0.y, v0.z, v0.w, v1.x },
            { v1.y, v1.z, v1.w, v2.x, v2.y },
            { v2.z, v2.w, v3.x, v3.y, v3.z },
            { v3.w, v4.x, v4.y, v4.z, v4.w },
        };
        int t[4] = { t01.x, t01.z, t23.x, t23.z };

#pragma unroll
        for (int s = 0; s < 4; ++s) sum += sample_term(p[s], t[s]);
    }

    // tail (n % 4), generic-safe; B = 8388608 has no tail
    for (long long i = (groups << 2) + tid; i < n; i += stride) {
        float p[NCLS];
#pragma unroll
        for (int j = 0; j < NCLS; ++j) p[j] = pred[NCLS * i + j];
        sum += sample_term(p, (int)truth[i]);
    }

    // deterministic block tree reduction in LDS
    __shared__ float red[THREADS];
    red[threadIdx.x] = sum;
    __syncthreads();
#pragma unroll
    for (int off = THREADS >> 1; off > 0; off >>= 1) {
        if ((int)threadIdx.x < off) red[threadIdx.x] += red[threadIdx.x + off];
        __syncthreads();
    }
    if (threadIdx.x == 0) partial[blockIdx.x] = red[0];
}

__global__ void __launch_bounds__(THREADS)
interclass_final_kernel(const float* __restrict__ partial, int nPartial,
                        float scale, float* __restrict__ out) {
    __shared__ float red[THREADS];
    float s = 0.0f;
    for (int i = threadIdx.x; i < nPartial; i += THREADS) s += partial[i];
    red[threadIdx.x] = s;
    __syncthreads();
#pragma unroll
    for (int off = THREADS >> 1; off > 0; off >>= 1) {
        if ((int)threadIdx.x < off) red[threadIdx.x] += red[threadIdx.x + off];
        __syncthreads();
    }
    if (threadIdx.x == 0) out[0] = red[0] * scale;
}

extern "C" void kernel_launch(void* const* d_in, const int* in_sizes, int n_in,
                              void* d_out, int out_size, void* d_ws, size_t ws_size,
                              hipStream_t stream) {
    const float*     pred  = (const float*)d_in[0];      // [B, 5] fp32
    const long long* truth = (const long long*)d_in[1];  // [B] int64
    const long long  n     = (long long)in_sizes[1];     // B

    float* partial = (float*)d_ws;                        // BLOCKS floats of scratch

    interclass_partial_kernel<<<BLOCKS, THREADS, 0, stream>>>(pred, truth, partial, n);
    interclass_final_kernel<<<1, THREADS, 0, stream>>>(partial, BLOCKS,
                                                       1.0f / (float)n, (float*)d_out);
}